// Signature_56203942035921
// MI455X (gfx1250) — compile-verified
//
#include <hip/hip_runtime.h>

typedef __attribute__((ext_vector_type(2))) float v2f;
typedef __attribute__((ext_vector_type(8))) float v8f;

#define C        8
#define LPATH    512
#define NPATH    64
#define NCHUNK   8
#define CHUNK    64            // segments per chunk (last chunk has 63)
#define L1SZ     8
#define L2SZ     64
#define L3SZ     512
#define L4SZ     4096
#define SIGSZ    (L1SZ + L2SZ + L3SZ + L4SZ)   // 4680
#define KA       20            // padded GEMM K (18 used in phase1, 17 in phase2)
#define ASTR     20            // A tile row stride in floats

// ---------------------------------------------------------------------------
// Phase 1: per (path, chunk) sequential Chen scan. Level-4 state lives in
// per-wave WMMA f32 accumulators (16 waves x one 16x16 tile of the 64x64
// level-4 matrix). Levels 1-3 live in LDS. Static tile structure (Kronecker
// zeros + K padding) is written once; the hot loop writes only dynamic entries.
// ---------------------------------------------------------------------------
__global__ __launch_bounds__(512)
void sig_chunk_kernel(const float* __restrict__ path, float* __restrict__ ws) {
    const int chunk = blockIdx.x;
    const int n     = blockIdx.y;
    const int tid   = threadIdx.x;
    const int t0    = chunk * CHUNK;
    const int rem   = (LPATH - 1) - t0;
    const int len   = rem < CHUNK ? rem : CHUNK;

    __shared__ float pb[(CHUNK + 1) * C];   // path points of this chunk
    __shared__ float dxb[CHUNK * C];        // per-segment increments
    __shared__ float s1[L1SZ];
    __shared__ float s2[L2SZ];
    __shared__ float s3[L3SZ];
    __shared__ float As[64 * ASTR];         // 64 x 20 (row major)
    __shared__ float Bs[KA * 64];           // 20 x 64 (row major)

    const int npts = (len + 1) * C;
    if (tid < npts)  pb[tid] = path[(size_t)n * LPATH * C + (size_t)t0 * C + tid];
    if (tid < L1SZ)  s1[tid] = 0.f;
    if (tid < L2SZ)  s2[tid] = 0.f;
    s3[tid] = 0.f;

    // ---- one-time static zeros: A off-diag of kron(s1,I8), A pad cols,
    //      B off-diag of kron(I8,dx), B pad rows ----
    #pragma unroll
    for (int e = tid; e < 64 * ASTR; e += 512) {
        int m = e / ASTR, j = e % ASTR;
        bool dyn = (j == (m & 7)) || (j >= 8 && j < 18);
        if (!dyn) As[e] = 0.f;
    }
    #pragma unroll
    for (int e = tid; e < KA * 64; e += 512) {
        int k = e >> 6, nn = e & 63;
        bool dyn = (k < 10) || (k < 18 && (nn >> 3) == (k - 10));
        if (!dyn) Bs[e] = 0.f;
    }
    __syncthreads();
    if (tid < len * C) dxb[tid] = pb[tid + C] - pb[tid];
    __syncthreads();

    // wave -> 16x16 tile of the 64x64 level-4 matrix
    const int lane = tid & 31;
    const int wv   = tid >> 5;
    const int Mb   = (wv >> 2) * 16;
    const int Nb   = (wv & 3) * 16;
    const int half = lane >> 4;
    const int lr   = lane & 15;
    const int aRow = (Mb + lr) * ASTR + 2 * half;  // contiguous v2f A fragment
    const int bCol = Nb + lr;

    v8f acc = {0.f, 0.f, 0.f, 0.f, 0.f, 0.f, 0.f, 0.f};

    const int i1 = tid >> 6, i2 = (tid >> 3) & 7, i3 = tid & 7;

    for (int s = 0; s < len; ++s) {
        const float* dx = &dxb[s * C];

        // ---- dynamic tile entries (read OLD sig) ----
        // A cols 10..17 = reshape(s3, 64x8); B rows 0..7 = reshape(e3, 8x64)
        As[(tid >> 3) * ASTR + 10 + (tid & 7)] = s3[tid];
        Bs[tid] = dx[tid >> 6] * dx[(tid >> 3) & 7] * dx[tid & 7] * (1.f / 6.f);

        // levels 1..3 Chen updates with OLD sig (into registers)
        const float d1 = dx[i1], d2 = dx[i2], d3 = dx[i3];
        float ns3 = s3[tid] + d1 * d2 * d3 * (1.f / 6.f)
                  + s1[i1] * (d2 * d3 * 0.5f)
                  + s2[(i1 << 3) | i2] * d3;
        float ns2 = 0.f, ns1 = 0.f;
        if (tid < L2SZ) {
            const float e2 = dx[tid >> 3] * dx[tid & 7];      // 2*e2 actually dx*dx
            As[tid * ASTR + (tid & 7)] = s1[tid >> 3];        // kron(s1, I8) diag
            As[tid * ASTR + 8]         = s2[tid];             // s2 (x) e2 term
            As[tid * ASTR + 9]         = e2 * (1.f / 12.f);   // e2/6 (e4 = e2(x)e2/6)
            Bs[8 * 64 + tid]           = e2 * 0.5f;           // e2
            Bs[9 * 64 + tid]           = e2 * 0.5f;           // e2
            Bs[(10 + (tid >> 3)) * 64 + tid] = dx[tid & 7];   // kron(I8, dx) diag
            ns2 = s2[tid] + e2 * 0.5f + s1[tid >> 3] * dx[tid & 7];
        }
        if (tid < L1SZ) ns1 = s1[tid] + dx[tid];
        __syncthreads();

        s3[tid] = ns3;
        if (tid < L2SZ) s2[tid] = ns2;
        if (tid < L1SZ) s1[tid] = ns1;

        // ---- level-4 accumulate: 5 x v_wmma_f32_16x16x4_f32 per tile ----
        #pragma unroll
        for (int k0 = 0; k0 < KA; k0 += 4) {
            v2f a = *(const v2f*)(As + aRow + k0);            // ds_load_b64
            v2f b;
            b.x = Bs[(k0 + 2 * half) * 64 + bCol];
            b.y = Bs[(k0 + 2 * half + 1) * 64 + bCol];
            acc = __builtin_amdgcn_wmma_f32_16x16x4_f32(
                false, a, false, b, (short)0, acc, false, false);
        }
        __syncthreads();
    }

    // ---- store chunk signature to workspace ----
    float* out = ws + ((size_t)n * NCHUNK + chunk) * SIGSZ;
    if (tid < L1SZ) out[tid] = s1[tid];
    if (tid < L2SZ) out[L1SZ + tid] = s2[tid];
    out[L1SZ + L2SZ + tid] = s3[tid];
    #pragma unroll
    for (int v = 0; v < 8; ++v) {
        int M = Mb + v + (half ? 8 : 0);
        out[L1SZ + L2SZ + L3SZ + M * 64 + Nb + lr] = acc[v];
    }
}

// ---------------------------------------------------------------------------
// Phase 2: per path, fold the 8 chunk signatures left-to-right with Chen's
// identity. Same Kronecker-factored GEMM for level 4 (K = 17 -> pad 20).
// b3 never touches LDS: each thread's global b3 element feeds both its Bs
// slot and its own level-3 update.
// ---------------------------------------------------------------------------
__global__ __launch_bounds__(512)
void sig_combine_kernel(const float* __restrict__ ws, float* __restrict__ outp) {
    const int n   = blockIdx.x;
    const int tid = threadIdx.x;

    __shared__ float a1[L1SZ];
    __shared__ float a2[L2SZ];
    __shared__ float a3[L3SZ];
    __shared__ float b1[L1SZ];
    __shared__ float b2[L2SZ];
    __shared__ float As[64 * ASTR];
    __shared__ float Bs[KA * 64];

    const int lane = tid & 31;
    const int wv   = tid >> 5;
    const int Mb   = (wv >> 2) * 16;
    const int Nb   = (wv & 3) * 16;
    const int half = lane >> 4;
    const int lr   = lane & 15;
    const int aRow = (Mb + lr) * ASTR + 2 * half;
    const int bCol = Nb + lr;
    const int i1 = tid >> 6, i2 = (tid >> 3) & 7, i3 = tid & 7;

    // one-time static zeros (A: off-diag kron + pad cols; B: off-diag kron + pad rows)
    #pragma unroll
    for (int e = tid; e < 64 * ASTR; e += 512) {
        int m = e / ASTR, j = e % ASTR;
        bool dyn = (j == (m & 7)) || (j >= 8 && j < 17);
        if (!dyn) As[e] = 0.f;
    }
    #pragma unroll
    for (int e = tid; e < KA * 64; e += 512) {
        int k = e >> 6, nn = e & 63;
        bool dyn = (k < 9) || (k < 17 && (nn >> 3) == (k - 9));
        if (!dyn) Bs[e] = 0.f;
    }

    const float* c0 = ws + (size_t)n * NCHUNK * SIGSZ;
    if (tid < L1SZ) a1[tid] = c0[tid];
    if (tid < L2SZ) a2[tid] = c0[L1SZ + tid];
    a3[tid] = c0[L1SZ + L2SZ + tid];
    v8f acc;
    #pragma unroll
    for (int v = 0; v < 8; ++v) {
        int M = Mb + v + (half ? 8 : 0);
        acc[v] = c0[L1SZ + L2SZ + L3SZ + M * 64 + Nb + lr];
    }

    for (int cidx = 1; cidx < NCHUNK; ++cidx) {
        const float* cb = ws + ((size_t)n * NCHUNK + cidx) * SIGSZ;
        __syncthreads();   // previous iteration's WMMA reads of As/Bs are done

        const float bb3 = cb[L1SZ + L2SZ + tid];   // b3 element, register only
        if (tid < L1SZ) b1[tid] = cb[tid];
        if (tid < L2SZ) b2[tid] = cb[L1SZ + tid];
        Bs[tid] = bb3;                                      // B rows 0..7
        As[(tid >> 3) * ASTR + 9 + (tid & 7)] = a3[tid];    // A cols 9..16
        if (tid < L2SZ) {
            As[tid * ASTR + (tid & 7)] = a1[tid >> 3];      // kron(a1, I8) diag
            As[tid * ASTR + 8]         = a2[tid];
        }
        __syncthreads();   // b1/b2 visible

        if (tid < L2SZ) {
            Bs[8 * 64 + tid]                = b2[tid];
            Bs[(9 + (tid >> 3)) * 64 + tid] = b1[tid & 7];  // kron(I8, b1) diag
        }
        // Chen levels 1..3 with OLD a
        float na3 = a3[tid] + bb3
                  + a1[i1] * b2[(i2 << 3) | i3]
                  + a2[(i1 << 3) | i2] * b1[i3];
        float na2 = 0.f, na1 = 0.f;
        if (tid < L2SZ) na2 = a2[tid] + b2[tid] + a1[tid >> 3] * b1[tid & 7];
        if (tid < L1SZ) na1 = a1[tid] + b1[tid];
        __syncthreads();   // Bs ready, old-a reads complete

        a3[tid] = na3;
        if (tid < L2SZ) a2[tid] = na2;
        if (tid < L1SZ) a1[tid] = na1;

        // level-4: acc += A*B (cross terms) + b4 (elementwise)
        #pragma unroll
        for (int k0 = 0; k0 < KA; k0 += 4) {
            v2f a = *(const v2f*)(As + aRow + k0);
            v2f b;
            b.x = Bs[(k0 + 2 * half) * 64 + bCol];
            b.y = Bs[(k0 + 2 * half + 1) * 64 + bCol];
            acc = __builtin_amdgcn_wmma_f32_16x16x4_f32(
                false, a, false, b, (short)0, acc, false, false);
        }
        #pragma unroll
        for (int v = 0; v < 8; ++v) {
            int M = Mb + v + (half ? 8 : 0);
            acc[v] += cb[L1SZ + L2SZ + L3SZ + M * 64 + Nb + lr];
        }
    }

    __syncthreads();
    float* o = outp + (size_t)n * SIGSZ;
    if (tid < L1SZ) o[tid] = a1[tid];
    if (tid < L2SZ) o[L1SZ + tid] = a2[tid];
    o[L1SZ + L2SZ + tid] = a3[tid];
    #pragma unroll
    for (int v = 0; v < 8; ++v) {
        int M = Mb + v + (half ? 8 : 0);
        o[L1SZ + L2SZ + L3SZ + M * 64 + Nb + lr] = acc[v];
    }
}

extern "C" void kernel_launch(void* const* d_in, const int* in_sizes, int n_in,
                              void* d_out, int out_size, void* d_ws, size_t ws_size,
                              hipStream_t stream) {
    const float* path = (const float*)d_in[0];
    float* out = (float*)d_out;
    float* ws  = (float*)d_ws;   // needs 64*8*4680*4 B = 9.6 MB of scratch

    dim3 g1(NCHUNK, NPATH, 1);
    hipLaunchKernelGGL(sig_chunk_kernel,   g1,          dim3(512), 0, stream, path, ws);
    hipLaunchKernelGGL(sig_combine_kernel, dim3(NPATH), dim3(512), 0, stream, ws, out);
}